// EncoderResGatedGraphConv_80015240725033
// MI455X (gfx1250) — compile-verified
//
#include <hip/hip_runtime.h>
#include <hip/hip_bf16.h>
#include <math.h>

// ---------------------------------------------------------------------------
// MI455X (gfx1250) implementation of a 2-layer ResGatedGraphConv GNN.
//
// Roofline: edge message passing moves ~3.2 GB (HBM-bound, ~140us @ 23.3TB/s);
// the 10 dense 100k x 64 x 64 GEMMs move ~0.5 GB and are memory-bound once the
// matrix math runs on the WMMA pipe -> fp32 end-to-end, V_WMMA_F32_16X16X4_F32.
//
// Data movement: the 128x64 fp32 node tile per workgroup is staged into LDS by
// the Tensor Data Mover (tensor_load_to_lds, TENSORcnt), with descriptor
// padding producing a bank-conflict-free 68-float row stride and tensor_dim1
// clipping giving zero-fill past row N. Each wave then register-blocks its
// whole 16x64 A tile (32 VGPRs) and batches B-fragment loads per column tile
// so the WMMA chain runs back-to-back instead of stalling per k-step.
// ---------------------------------------------------------------------------

typedef __attribute__((ext_vector_type(2))) float v2f;
typedef __attribute__((ext_vector_type(8))) float v8f;
typedef __attribute__((ext_vector_type(4))) unsigned int u32x4;
typedef __attribute__((ext_vector_type(4))) int i32x4;
typedef __attribute__((ext_vector_type(8))) int i32x8;

#define NNODES 100000    // multiple of 16 -> full 16-row WMMA tiles everywhere
#define NEDGES 1600000
#define DIM 64
#define ROWS_PER_WAVE 16
#define WAVES_PER_BLOCK 8
#define ROWS_PER_BLOCK (ROWS_PER_WAVE * WAVES_PER_BLOCK)   // 128
#define LDS_STRIDE 68   // 64 + 4 pad -> conflict-free ds_load_b64 A fragments

// ---------------------------------------------------------------------------
// TDM: DMA one 128x64 fp32 tile (rows block_row0..+127 of X[N][64]) into LDS
// at lds_byte_off, padded to LDS_STRIDE floats per row. Rows >= nrows read as
// zero via tensor_dim1 OOB clipping. Issued by one wave; that wave waits on
// TENSORcnt, then the workgroup barrier publishes the tile.
// ---------------------------------------------------------------------------
__device__ __forceinline__ void tdm_stage_block(
    const float* __restrict__ X, unsigned int lds_byte_off,
    int block_row0, int nrows)
{
  const unsigned long long gaddr =
      (unsigned long long)(uintptr_t)(X + (size_t)block_row0 * DIM);

  u32x4 g0;
  g0.x = 1u;                                           // count=1, user mode
  g0.y = lds_byte_off;                                 // LDS dest (bytes)
  g0.z = (unsigned int)gaddr;                          // global_addr[31:0]
  g0.w = ((unsigned int)(gaddr >> 32) & 0x01FFFFFFu)   // global_addr[56:32]
         | (2u << 30);                                 // type = 2 ("image")

  i32x8 g1;
  g1[0] = (int)((2u << 16)      // data_size = 2 -> 4 bytes
              | (1u << 20)      // pad_enable
              | (5u << 22)      // pad_interval code 5 -> every 64 DWORDs
              | (3u << 25));    // pad_amount   code 3 -> 4 DWORDs
  g1[1] = (int)(64u << 16);                             // tensor_dim0 = 64
  g1[2] = (int)(((unsigned)nrows & 0xFFFFu) << 16);     // tensor_dim1 lo
  g1[3] = (int)(((unsigned)nrows >> 16) | (64u << 16)); // tensor_dim1 hi | tile_dim0=64
  g1[4] = (int)ROWS_PER_BLOCK;                          // tile_dim1 = 128
  g1[5] = (int)DIM;                                     // tensor_dim0_stride = 64
  g1[6] = 0;
  g1[7] = 0;

  const i32x4 zero4 = {0, 0, 0, 0};
  const i32x8 zero8 = {0, 0, 0, 0, 0, 0, 0, 0};
  // amdgpu-toolchain 6-arg form: (g0, g1, g2, g3, extra, cpol).
  __builtin_amdgcn_tensor_load_to_lds(g0, g1, zero4, zero4, zero8, 0);
}

// ---------------------------------------------------------------------------
// Load this wave's entire 16x64 A tile from LDS into 16 v2f fragments
// (32 VGPRs / lane = whole tile across the wave); reused for every weight
// matrix and every output column tile.
// A fragment (16x4 f32): lanes 0-15 hold M=lane, K={kk,kk+1}; lanes 16-31
// hold M=lane-16, K={kk+2,kk+3}  (ISA 7.12.2, 32-bit A 16x4 layout).
// ---------------------------------------------------------------------------
__device__ __forceinline__ void load_a_frags(
    const float* __restrict__ lds_tile, int lane, v2f afrag[16])
{
  const int lh = lane >> 4;
  const int ll = lane & 15;
  const float* base = &lds_tile[ll * LDS_STRIDE + lh * 2];
#pragma unroll
  for (int kt = 0; kt < 16; ++kt) {
    afrag[kt] = *(const v2f*)&base[kt * 4];   // ds_load_b64, conflict-free
  }
}

// ---------------------------------------------------------------------------
// 16x64 output tile = A(regs) * W(64x64) on the WMMA pipe.
// Per column tile: batch-load all 16 B fragments (loads pipeline as one
// clause), then issue 16 back-to-back v_wmma_f32_16x16x4_f32 accumulating
// D->C. Stores unguarded: nrows % 16 == 0 so every live tile is full.
// ---------------------------------------------------------------------------
__device__ __forceinline__ void wave_gemm_16x64(
    const v2f afrag[16],
    const float* __restrict__ W,         // [64][64] row-major (din x dout)
    float* __restrict__ out,             // [N][64]
    int row0, int lane,
    const float* __restrict__ bias,      // nullptr -> no bias
    bool relu)
{
  const int lh = lane >> 4;   // half-wave: 0 or 1
  const int ll = lane & 15;

#pragma unroll
  for (int nt = 0; nt < 4; ++nt) {
    const int n0  = nt * 16;
    const int col = n0 + ll;

    // Batch the 32 B-fragment loads for this column tile.
    v2f bf[16];
#pragma unroll
    for (int kt = 0; kt < 16; ++kt) {
      const int kk = kt * 4 + lh * 2;     // this lane's K pair
      bf[kt].x = W[(kk + 0) * DIM + col];
      bf[kt].y = W[(kk + 1) * DIM + col];
    }

    v8f acc = {0.f, 0.f, 0.f, 0.f, 0.f, 0.f, 0.f, 0.f};
#pragma unroll
    for (int kt = 0; kt < 16; ++kt) {
      acc = __builtin_amdgcn_wmma_f32_16x16x4_f32(
          false, afrag[kt], false, bf[kt], (short)0, acc, false, false);
    }

    const float bv = bias ? bias[col] : 0.f;
#pragma unroll
    for (int r = 0; r < 8; ++r) {
      const int grow = row0 + r + lh * 8;
      float o = acc[r] + bv;
      if (relu) o = fmaxf(o, 0.f);
      out[grow * DIM + col] = o;   // full tile: no per-row predication
    }
  }
}

// ---------------------------------------------------------------------------
// Fused K/Q/V/S projections: K=x@Wk, Q=x@Wq, V=x@Wv, AGG = x@Ws + b
// (AGG seeded with the skip+bias term; edge kernel atomically adds messages).
// ---------------------------------------------------------------------------
__global__ __launch_bounds__(256) void kqvs_kernel(
    const float* __restrict__ X,
    const float* __restrict__ Wk, const float* __restrict__ Wq,
    const float* __restrict__ Wv, const float* __restrict__ Ws,
    const float* __restrict__ b,
    float* __restrict__ K, float* __restrict__ Q,
    float* __restrict__ V, float* __restrict__ AGG,
    int nrows)
{
  __shared__ float tile[WAVES_PER_BLOCK * ROWS_PER_WAVE * LDS_STRIDE];
  const int tid  = threadIdx.x;
  const int wave = tid >> 5;
  const int lane = tid & 31;
  const int block_row0 = blockIdx.x * ROWS_PER_BLOCK;
  const int row0 = block_row0 + wave * ROWS_PER_WAVE;

  if (wave == 0) {   // one TDM DMA stages the whole block's tile
    tdm_stage_block(X, (unsigned int)(uintptr_t)(void*)tile, block_row0, nrows);
    __builtin_amdgcn_s_wait_tensorcnt(0);
  }
  __syncthreads();

  if (row0 < nrows) {   // wave-uniform; EXEC stays all-ones around WMMA
    const float* my = &tile[wave * ROWS_PER_WAVE * LDS_STRIDE];
    v2f afrag[16];
    load_a_frags(my, lane, afrag);   // whole A tile in regs, reused 4x below
    wave_gemm_16x64(afrag, Wk, K,   row0, lane, nullptr, false);
    wave_gemm_16x64(afrag, Wq, Q,   row0, lane, nullptr, false);
    wave_gemm_16x64(afrag, Wv, V,   row0, lane, nullptr, false);
    wave_gemm_16x64(afrag, Ws, AGG, row0, lane, b,       false);
  }
}

// ---------------------------------------------------------------------------
// Linear + bias + ReLU:  O = relu(X @ W + b)
// ---------------------------------------------------------------------------
__global__ __launch_bounds__(256) void linear_relu_kernel(
    const float* __restrict__ X, const float* __restrict__ W,
    const float* __restrict__ b, float* __restrict__ O, int nrows)
{
  __shared__ float tile[WAVES_PER_BLOCK * ROWS_PER_WAVE * LDS_STRIDE];
  const int tid  = threadIdx.x;
  const int wave = tid >> 5;
  const int lane = tid & 31;
  const int block_row0 = blockIdx.x * ROWS_PER_BLOCK;
  const int row0 = block_row0 + wave * ROWS_PER_WAVE;

  if (wave == 0) {
    tdm_stage_block(X, (unsigned int)(uintptr_t)(void*)tile, block_row0, nrows);
    __builtin_amdgcn_s_wait_tensorcnt(0);
  }
  __syncthreads();

  if (row0 < nrows) {
    const float* my = &tile[wave * ROWS_PER_WAVE * LDS_STRIDE];
    v2f afrag[16];
    load_a_frags(my, lane, afrag);
    wave_gemm_16x64(afrag, W, O, row0, lane, b, true);
  }
}

// ---------------------------------------------------------------------------
// Edge messages: AGG[dst] += sigmoid(K[dst] + Q[src]) * V[src]
// 16 lanes per edge, float4 per lane (256B coalesced row reads),
// global_atomic_add_f32 scatter (L2 atomic units absorb the contention).
// ---------------------------------------------------------------------------
__global__ __launch_bounds__(256) void edge_gate_kernel(
    const float* __restrict__ K, const float* __restrict__ Q,
    const float* __restrict__ V,
    const int* __restrict__ src, const int* __restrict__ dst,
    float* __restrict__ AGG, int nedges)
{
  const long long tid = (long long)blockIdx.x * 256 + threadIdx.x;
  const int e = (int)(tid >> 4);
  if (e >= nedges) return;
  const int c = (int)(tid & 15) << 2;

  const int s = src[e];
  const int d = dst[e];

  const float4 kv = *(const float4*)&K[d * DIM + c];
  const float4 qv = *(const float4*)&Q[s * DIM + c];
  const float4 vv = *(const float4*)&V[s * DIM + c];

  const float g0 = 1.f / (1.f + __expf(-(kv.x + qv.x)));
  const float g1 = 1.f / (1.f + __expf(-(kv.y + qv.y)));
  const float g2 = 1.f / (1.f + __expf(-(kv.z + qv.z)));
  const float g3 = 1.f / (1.f + __expf(-(kv.w + qv.w)));

  float* out = &AGG[d * DIM + c];
  atomicAdd(out + 0, g0 * vv.x);
  atomicAdd(out + 1, g1 * vv.y);
  atomicAdd(out + 2, g2 * vv.z);
  atomicAdd(out + 3, g3 * vv.w);
}

// ---------------------------------------------------------------------------
// Host-side launch: 2 x (KQVS -> edge scatter -> linear+relu)
// ---------------------------------------------------------------------------
extern "C" void kernel_launch(void* const* d_in, const int* in_sizes, int n_in,
                              void* d_out, int out_size, void* d_ws, size_t ws_size,
                              hipStream_t stream) {
  (void)in_sizes; (void)n_in; (void)out_size; (void)ws_size;

  const float* x   = (const float*)d_in[0];
  const float* Wk1 = (const float*)d_in[1];
  const float* Wq1 = (const float*)d_in[2];
  const float* Wv1 = (const float*)d_in[3];
  const float* Ws1 = (const float*)d_in[4];
  const float* b1  = (const float*)d_in[5];
  const float* Wl1 = (const float*)d_in[6];
  const float* bl1 = (const float*)d_in[7];
  const float* Wk2 = (const float*)d_in[8];
  const float* Wq2 = (const float*)d_in[9];
  const float* Wv2 = (const float*)d_in[10];
  const float* Ws2 = (const float*)d_in[11];
  const float* b2  = (const float*)d_in[12];
  const float* Wl2 = (const float*)d_in[13];
  const float* bl2 = (const float*)d_in[14];
  const int*   ei  = (const int*)d_in[15];
  const int* src = ei;            // edge_index[0, :]
  const int* dst = ei + NEDGES;   // edge_index[1, :]

  float* ws = (float*)d_ws;
  const size_t NF = (size_t)NNODES * DIM;       // 6.4M floats (25.6 MB)
  float* K   = ws;
  float* Q   = ws + NF;
  float* V   = ws + 2 * NF;
  float* AGG = ws + 3 * NF;
  float* H   = ws + 4 * NF;
  float* out = (float*)d_out;

  const dim3 tb(256);
  const dim3 gemm_grid((NNODES + ROWS_PER_BLOCK - 1) / ROWS_PER_BLOCK);  // 782
  const dim3 edge_grid((unsigned)(((long long)NEDGES * 16 + 255) / 256)); // 100000

  // ---- layer 1 ----
  kqvs_kernel<<<gemm_grid, tb, 0, stream>>>(x, Wk1, Wq1, Wv1, Ws1, b1,
                                            K, Q, V, AGG, NNODES);
  edge_gate_kernel<<<edge_grid, tb, 0, stream>>>(K, Q, V, src, dst, AGG, NEDGES);
  linear_relu_kernel<<<gemm_grid, tb, 0, stream>>>(AGG, Wl1, bl1, H, NNODES);

  // ---- layer 2 ----
  kqvs_kernel<<<gemm_grid, tb, 0, stream>>>(H, Wk2, Wq2, Wv2, Ws2, b2,
                                            K, Q, V, AGG, NNODES);
  edge_gate_kernel<<<edge_grid, tb, 0, stream>>>(K, Q, V, src, dst, AGG, NEDGES);
  linear_relu_kernel<<<gemm_grid, tb, 0, stream>>>(AGG, Wl2, bl2, out, NNODES);
}